// ListMLELoss_enhanced_38336878084752
// MI455X (gfx1250) — compile-verified
//
#include <hip/hip_runtime.h>
#include <stdint.h>

#define NCOL 2048
#define NT   512
#define TOLR 0.05f
#define EPSV 1e-10f
#define NOISE_SCALE 1e-6f

typedef __attribute__((ext_vector_type(2))) float v2f;
typedef __attribute__((ext_vector_type(8))) float v8f;

#if defined(__has_builtin)
#if __has_builtin(__builtin_amdgcn_wmma_f32_16x16x4_f32)
#define HAVE_WMMA4 1
#endif
#endif

__device__ __forceinline__ float hash01(uint32_t x) {
  x ^= x >> 16; x *= 0x7feb352dU;
  x ^= x >> 15; x *= 0x846ca68bU;
  x ^= x >> 16;
  return (float)(x >> 8) * (1.0f / 16777216.0f);
}

// In-LDS bitonic sort, descending by value, ties broken by ascending index
// (emulates a stable descending argsort).
__device__ __forceinline__ void bitonic_desc(float* v, int* ix, int tid) {
  for (unsigned k = 2; k <= NCOL; k <<= 1) {
    for (unsigned j = k >> 1; j > 0; j >>= 1) {
      __syncthreads();
#pragma unroll
      for (int pp = 0; pp < (NCOL / 2) / NT; ++pp) {
        unsigned p = (unsigned)tid + (unsigned)pp * NT;
        unsigned i = ((p & ~(j - 1u)) << 1) | (p & (j - 1u));
        unsigned l = i + j;
        float a = v[i], b = v[l];
        int ia = ix[i], ib = ix[l];
        bool aAfter = (a < b) || (a == b && ia > ib);  // a belongs after b (desc)
        bool desc = ((i & k) == 0);
        if (desc ? aAfter : !aAfter) {
          v[i] = b; v[l] = a; ix[i] = ib; ix[l] = ia;
        }
      }
    }
  }
  __syncthreads();
}

extern "C" __global__ void __launch_bounds__(NT)
listmle_row_kernel(const float* __restrict__ pred,
                   const float* __restrict__ gt,
                   float* __restrict__ rowLoss) {
  __shared__ float sv[NCOL];          // gt values -> noisy gt -> ordered preds
  __shared__ float se[NCOL];          // exp values -> suffix sums
  __shared__ int   sidx[NCOL];        // argsort payload
  __shared__ float red[NT];
  __shared__ float chunkExcl[NCOL / 16];

  const int tid = threadIdx.x;
  const int row = blockIdx.x;
  const float* gtr = gt + (size_t)row * NCOL;
  const float* prr = pred + (size_t)row * NCOL;

  // --- Stage gt row into LDS with CDNA5 async global->LDS copies (ASYNCcnt) ---
  for (int i = tid; i < NCOL; i += NT) {
    uint32_t ldsAddr = (uint32_t)(uintptr_t)(&sv[i]);
    const float* ga = gtr + i;
    asm volatile("global_load_async_to_lds_b32 %0, %1, off"
                 :: "v"(ldsAddr), "v"(ga) : "memory");
    sidx[i] = i;
  }
  asm volatile("s_wait_asynccnt 0x0" ::: "memory");
  __syncthreads();

  // --- Sort 1: descending gt with index payload ---
  bitonic_desc(sv, sidx, tid);

  // --- Tie-group noise: element is in a group of size>1 iff within tolerance
  //     of a sorted neighbor. Read phase -> barrier -> write phase. ---
  float nbuf[NCOL / NT];
#pragma unroll
  for (int pp = 0; pp < NCOL / NT; ++pp) {
    int i = tid + pp * NT;
    float c = sv[i];
    bool tied = false;
    if (i > 0        && fabsf(sv[i - 1] - c) <= TOLR) tied = true;
    if (i < NCOL - 1 && fabsf(c - sv[i + 1]) <= TOLR) tied = true;
    nbuf[pp] = tied
        ? c + hash01(((uint32_t)row * (uint32_t)NCOL + (uint32_t)i) * 2654435761u) * NOISE_SCALE
        : c;
  }
  __syncthreads();
#pragma unroll
  for (int pp = 0; pp < NCOL / NT; ++pp) sv[tid + pp * NT] = nbuf[pp];

  // --- Sort 2: re-sort noisy values (bitonic_desc begins with a barrier) ---
  bitonic_desc(sv, sidx, tid);

  // --- Gather ordered predictions into sv; block max reduction ---
  float lmax = -INFINITY;
#pragma unroll
  for (int pp = 0; pp < NCOL / NT; ++pp) {
    int i = tid + pp * NT;
    float p = prr[sidx[i]];
    sv[i] = p;                 // sv now holds ordered predictions
    lmax = fmaxf(lmax, p);
  }
  red[tid] = lmax;
  __syncthreads();
  for (int s = NT / 2; s > 0; s >>= 1) {
    if (tid < s) red[tid] = fmaxf(red[tid], red[tid + s]);
    __syncthreads();
  }
  const float m = red[0];

  // --- e = exp(op - m) ---
#pragma unroll
  for (int pp = 0; pp < NCOL / NT; ++pp) {
    int i = tid + pp * NT;
    se[i] = expf(sv[i] - m);
  }
  __syncthreads();

  // --- Inclusive suffix sums within 16-element chunks via WMMA:
  //     Y = T * X, T[i][j] = (j>=i) ? 1 : 0 (upper-triangular ones),
  //     X column c = chunk c. T split into four 16x4 slabs, 4 chained
  //     V_WMMA_F32_16X16X4_F32 per wave; wave w owns elements [w*256, w*256+255].
  const int wave = tid >> 5;
  const int lane = tid & 31;
#ifdef HAVE_WMMA4
  if (wave < 8) {
    const int segBase = wave * 256;
    const int nloc = lane & 15;       // A: row M; B: column N (chunk index)
    const int hi = (lane >> 4) & 1;   // lanes 16-31 carry K+2 / K+3
    v8f acc = {0.f, 0.f, 0.f, 0.f, 0.f, 0.f, 0.f, 0.f};
#pragma unroll
    for (int k = 0; k < 4; ++k) {
      const int k0 = 4 * k + (hi ? 2 : 0);
      const int k1 = 4 * k + (hi ? 3 : 1);
      v2f A, B;
      A.x = (k0 >= nloc) ? 1.0f : 0.0f;
      A.y = (k1 >= nloc) ? 1.0f : 0.0f;
      B.x = se[segBase + nloc * 16 + k0];
      B.y = se[segBase + nloc * 16 + k1];
      acc = __builtin_amdgcn_wmma_f32_16x16x4_f32(
          false, A, false, B, (short)0, acc, false, false);
    }
#pragma unroll
    for (int r = 0; r < 8; ++r) {
      // D vgpr r: lanes 0-15 -> M=r, lanes 16-31 -> M=r+8; N = nloc
      se[segBase + nloc * 16 + (r + hi * 8)] = acc[r];
    }
  }
#else
  for (int c = tid; c < NCOL / 16; c += NT) {
    float run = 0.f;
    for (int j2 = 15; j2 >= 0; --j2) { run += se[c * 16 + j2]; se[c * 16 + j2] = run; }
  }
#endif
  __syncthreads();

  // --- Exclusive suffix over the 128 chunk totals (chunk total = se[c*16]) ---
  if (tid == 0) {
    float run = 0.f;
    for (int c = NCOL / 16 - 1; c >= 0; --c) {
      chunkExcl[c] = run;
      run += se[c * 16];
    }
  }
  __syncthreads();

  // --- Row loss: -(op - m - log(max(suffix, EPS))) summed ---
  float accLoss = 0.f;
#pragma unroll
  for (int pp = 0; pp < NCOL / NT; ++pp) {
    int i = tid + pp * NT;
    float suf = se[i] + chunkExcl[i >> 4];
    suf = fmaxf(suf, EPSV);
    accLoss += -(sv[i] - m - logf(suf));
  }
  red[tid] = accLoss;
  __syncthreads();
  for (int s = NT / 2; s > 0; s >>= 1) {
    if (tid < s) red[tid] += red[tid + s];
    __syncthreads();
  }
  if (tid == 0) rowLoss[row] = red[0];
}

extern "C" __global__ void __launch_bounds__(1024)
listmle_reduce_kernel(const float* __restrict__ rowLoss,
                      float* __restrict__ out, int rows) {
  __shared__ float red[1024];
  const int t = threadIdx.x;
  float s = 0.f;
  for (int i = t; i < rows; i += 1024) s += rowLoss[i];
  red[t] = s;
  __syncthreads();
  for (int st = 512; st > 0; st >>= 1) {
    if (t < st) red[t] += red[t + st];
    __syncthreads();
  }
  if (t == 0) out[0] = red[0] / (float)rows;
}

extern "C" void kernel_launch(void* const* d_in, const int* in_sizes, int n_in,
                              void* d_out, int out_size, void* d_ws, size_t ws_size,
                              hipStream_t stream) {
  const float* pred = (const float*)d_in[0];
  const float* gt   = (const float*)d_in[1];
  float* out = (float*)d_out;
  float* rowLoss = (float*)d_ws;                 // rows * 4 bytes of scratch

  const int rows = in_sizes[1] / NCOL;           // 4096 for the reference shapes

  listmle_row_kernel<<<rows, NT, 0, stream>>>(pred, gt, rowLoss);
  listmle_reduce_kernel<<<1, 1024, 0, stream>>>(rowLoss, out, rows);
}